// GraphConvolutionlayer_41180146434554
// MI455X (gfx1250) — compile-verified
//
#include <hip/hip_runtime.h>
#include <hip/hip_bf16.h>

// ---------------------------------------------------------------------------
// GCN layer for MI455X (gfx1250, wave32):
//   support = x @ W            -- fp32 WMMA (V_WMMA_F32_16X16X4_F32), exact
//   out     = bias + scatter_add(edge_vals * support[edge_cols] -> edge_rows)
// ---------------------------------------------------------------------------

typedef float v2f __attribute__((ext_vector_type(2)));
typedef float v8f __attribute__((ext_vector_type(8)));

#define N_NODES 50000
#define F_DIM   128

// ---------------------------------------------------------------------------
// Kernel 1: support = X(50000x128) @ W(128x128), fp32 WMMA 16x16x4.
// Block = 256 threads = 8 waves. blockIdx.x -> 16-row slab, wave -> 16-col tile.
// ---------------------------------------------------------------------------
__global__ __launch_bounds__(256)
void gcn_gemm_wmma(const float* __restrict__ X,
                   const float* __restrict__ W,
                   float* __restrict__ S) {
    const int wave = threadIdx.x >> 5;      // 0..7 : column tile
    const int lane = threadIdx.x & 31;
    const int half = lane >> 4;             // 0 or 1 (lane group)
    const int mr   = lane & 15;             // A-row / B-col / D-col index
    const int kb   = half * 2;              // K sub-offset within group of 4

    const int m0 = blockIdx.x * 16;
    const int n0 = wave * 16;

    v8f acc = {};

    const float* arow = X + (size_t)(m0 + mr) * F_DIM;   // A: row m0+mr
    const float* bcol = W + n0 + mr;                     // B: col n0+mr

    #pragma unroll 4
    for (int k = 0; k < F_DIM; k += 4) {
        // A fragment (16x4 f32): v0 = A[m][k+kb], v1 = A[m][k+kb+1]
        v2f a = *(const v2f*)(arow + k + kb);            // 8B-aligned float2
        // B fragment (4x16 f32): v0 = W[k+kb][n], v1 = W[k+kb+1][n]
        v2f b;
        b.x = bcol[(size_t)(k + kb) * F_DIM];
        b.y = bcol[(size_t)(k + kb + 1) * F_DIM];
        // 8 args: (neg_a, A, neg_b, B, c_mod, C, reuse_a, reuse_b)
        acc = __builtin_amdgcn_wmma_f32_16x16x4_f32(
            false, a, false, b, (short)0, acc, false, false);
    }

    // D layout: VGPR i -> D[M = i + 8*half][N = mr]
    float* sp = S + (size_t)(m0 + 8 * half) * F_DIM + n0 + mr;
    #pragma unroll
    for (int i = 0; i < 8; ++i)
        sp[(size_t)i * F_DIM] = acc[i];
}

// ---------------------------------------------------------------------------
// Kernel 2: out[n][f] = bias[f]   (harness poisons d_out; fold bias here)
// ---------------------------------------------------------------------------
__global__ __launch_bounds__(256)
void gcn_init_out(const float* __restrict__ bias,
                  float* __restrict__ out,
                  int total) {
    int i = blockIdx.x * blockDim.x + threadIdx.x;
    if (i < total)
        out[i] = bias[i & (F_DIM - 1)];
}

// ---------------------------------------------------------------------------
// Kernel 3: one wave per edge. Wave-uniform r/c/v (scalar broadcast);
// each lane handles 4 features: float4 gather from support (L2-resident),
// 4x global_atomic_add_f32 scatter into out (L2-resident).
// ---------------------------------------------------------------------------
__global__ __launch_bounds__(256)
void gcn_spmm_scatter(const float* __restrict__ S,
                      const float* __restrict__ vals,
                      const int* __restrict__ rows,
                      const int* __restrict__ cols,
                      float* __restrict__ out,
                      int nEdges) {
    const int wavesPerBlock = blockDim.x >> 5;
    const int edge = blockIdx.x * wavesPerBlock + (threadIdx.x >> 5);
    if (edge >= nEdges) return;
    const int lane = threadIdx.x & 31;

    const int   r = rows[edge];
    const int   c = cols[edge];
    const float v = vals[edge];

    const float4 s = *(const float4*)(S + (size_t)c * F_DIM + lane * 4);
    float* o = out + (size_t)r * F_DIM + lane * 4;

    atomicAdd(o + 0, v * s.x);
    atomicAdd(o + 1, v * s.y);
    atomicAdd(o + 2, v * s.z);
    atomicAdd(o + 3, v * s.w);
}

// ---------------------------------------------------------------------------
// Launch
// ---------------------------------------------------------------------------
extern "C" void kernel_launch(void* const* d_in, const int* in_sizes, int n_in,
                              void* d_out, int out_size, void* d_ws, size_t ws_size,
                              hipStream_t stream) {
    const float* x      = (const float*)d_in[0];   // [50000,128]
    const float* weight = (const float*)d_in[1];   // [128,128]
    const float* bias   = (const float*)d_in[2];   // [128]
    const float* evals  = (const float*)d_in[3];   // [E]
    const int*   erows  = (const int*)d_in[4];     // [E]
    const int*   ecols  = (const int*)d_in[5];     // [E]
    float*       out    = (float*)d_out;           // [50000,128]

    const int nEdges = in_sizes[3];
    const int nNodes = in_sizes[0] / F_DIM;

    float* support = (float*)d_ws;                 // 50000*128*4 = 25.6 MB

    // 1) GEMM: 16-row slabs, 8 col-tiles per block (one per wave)
    dim3 gemmGrid((nNodes + 15) / 16);
    gcn_gemm_wmma<<<gemmGrid, 256, 0, stream>>>(x, weight, support);

    // 2) out = bias (broadcast)
    const int total = nNodes * F_DIM;
    gcn_init_out<<<(total + 255) / 256, 256, 0, stream>>>(bias, out, total);

    // 3) scatter-add: one wave per edge, 8 edges per block
    const int edgeBlocks = (nEdges + 7) / 8;
    gcn_spmm_scatter<<<edgeBlocks, 256, 0, stream>>>(
        support, evals, erows, ecols, out, nEdges);
}